// PolyModel_34394098106515
// MI455X (gfx1250) — compile-verified
//
#include <hip/hip_runtime.h>

// CDNA5 (gfx1250) wave32 WMMA types
typedef __attribute__((ext_vector_type(2))) float v2f;   // A/B operand: 16x4 / 4x16 f32 = 2 VGPRs
typedef __attribute__((ext_vector_type(8))) float v8f;   // C/D operand: 16x16 f32 = 8 VGPRs

// One 16-element polynomial evaluation chain on the matrix pipe:
//   D[m][n] = b + sum_{k=0..7} (w_k/k!) * xa_n^k        (replicated over m)
// A holds coefficients (row-invariant), B holds powers of x.
//   B VGPR0 = {K0 | K2} by lane half, VGPR1 = {K1 | K3}
//   second WMMA shifts K by 4 via multiplying B by x^4.
__device__ __forceinline__ v8f poly_chain(float xa, bool hi,
                                          v2f Alo, v2f Ahi, v8f C) {
  float s = xa * xa;      // x^2
  float q = s * s;        // x^4
  v2f B, Bq;
  B[0]  = hi ? s : 1.0f;  // K0 | K2
  B[1]  = B[0] * xa;      // K1 | K3
  Bq[0] = B[0] * q;       // K4 | K6
  Bq[1] = B[1] * q;       // K5 | K7
  v8f D = __builtin_amdgcn_wmma_f32_16x16x4_f32(
      /*neg_a=*/false, Alo, /*neg_b=*/false, B,  (short)0, C, false, false);
  D = __builtin_amdgcn_wmma_f32_16x16x4_f32(
      /*neg_a=*/false, Ahi, /*neg_b=*/false, Bq, (short)0, D, false, false);
  return D;
}

__global__ void PolyModel_wmma_kernel(const float* __restrict__ X,
                                      const float* __restrict__ w,
                                      const float* __restrict__ b,
                                      float* __restrict__ out,
                                      int n) {
  const int  lane = threadIdx.x & 31;
  const bool hi   = lane >= 16;

  // Fold 1/k! into the weights (uniform scalar loads; tiny, L2-resident).
  const float c0 = w[0];
  const float c1 = w[1];
  const float c2 = w[2] * 0.5f;
  const float c3 = w[3] * (1.0f / 6.0f);
  const float c4 = w[4] * (1.0f / 24.0f);
  const float c5 = w[5] * (1.0f / 120.0f);
  const float c6 = w[6] * (1.0f / 720.0f);
  const float c7 = w[7] * (1.0f / 5040.0f);
  const float bv = b[0];

  // A-matrix (16x4 f32): VGPR0 = {K0 | K2}, VGPR1 = {K1 | K3}; rows identical.
  v2f Alo, Ahi;
  Alo[0] = hi ? c2 : c0;
  Alo[1] = hi ? c3 : c1;
  Ahi[0] = hi ? c6 : c4;
  Ahi[1] = hi ? c7 : c5;

  v8f C;
  #pragma unroll
  for (int i = 0; i < 8; ++i) C[i] = bv;   // bias pre-loaded into accumulator

  const long tid         = (long)blockIdx.x * blockDim.x + threadIdx.x;
  const long wave        = tid >> 5;
  const long total_waves = ((long)gridDim.x * blockDim.x) >> 5;
  const int  nsel        = lane & 15;

  // Main loop: 128 elements / wave / iteration, wave-uniform bounds (EXEC all-1s).
  #pragma unroll 1
  for (long base = wave * 128; base + 127 < (long)n; base += total_waves * 128) {
    float4 x4 = ((const float4*)(X + base))[lane];   // global_load_b128

    // Broadcast: chain (c,g) column n needs component c of lane g*16+n.
    float4 lo4, hi4;
    lo4.x = __shfl(x4.x, nsel,      32);
    lo4.y = __shfl(x4.y, nsel,      32);
    lo4.z = __shfl(x4.z, nsel,      32);
    lo4.w = __shfl(x4.w, nsel,      32);
    hi4.x = __shfl(x4.x, nsel + 16, 32);
    hi4.y = __shfl(x4.y, nsel + 16, 32);
    hi4.z = __shfl(x4.z, nsel + 16, 32);
    hi4.w = __shfl(x4.w, nsel + 16, 32);

    // 8 chains of 16 elements: elem(c,g,n) = base + 4*(g*16+n) + c
    v8f D0g0 = poly_chain(lo4.x, hi, Alo, Ahi, C);
    v8f D0g1 = poly_chain(hi4.x, hi, Alo, Ahi, C);
    v8f D1g0 = poly_chain(lo4.y, hi, Alo, Ahi, C);
    v8f D1g1 = poly_chain(hi4.y, hi, Alo, Ahi, C);
    v8f D2g0 = poly_chain(lo4.z, hi, Alo, Ahi, C);
    v8f D2g1 = poly_chain(hi4.z, hi, Alo, Ahi, C);
    v8f D3g0 = poly_chain(lo4.w, hi, Alo, Ahi, C);
    v8f D3g1 = poly_chain(hi4.w, hi, Alo, Ahi, C);

    // Lane L holds column n = L&15 in every D VGPR; picking g = L>>4 makes
    // lane L's results exactly elements base+4L .. base+4L+3 -> b128 store.
    float4 o;
    o.x = hi ? D0g1[0] : D0g0[0];
    o.y = hi ? D1g1[0] : D1g0[0];
    o.z = hi ? D2g1[0] : D2g0[0];
    o.w = hi ? D3g1[0] : D3g0[0];
    ((float4*)(out + base))[lane] = o;                // global_store_b128
  }

  // Scalar Horner tail for N % 128 != 0 (divergence-safe: no WMMA here).
  const long tail_start = (long)n & ~127L;
  for (long i = tail_start + tid; i < (long)n; i += (long)gridDim.x * blockDim.x) {
    float x = X[i];
    float y = c7;
    y = fmaf(y, x, c6);
    y = fmaf(y, x, c5);
    y = fmaf(y, x, c4);
    y = fmaf(y, x, c3);
    y = fmaf(y, x, c2);
    y = fmaf(y, x, c1);
    y = fmaf(y, x, c0);
    out[i] = y + bv;
  }
}

extern "C" void kernel_launch(void* const* d_in, const int* in_sizes, int n_in,
                              void* d_out, int out_size, void* d_ws, size_t ws_size,
                              hipStream_t stream) {
  const float* X = (const float*)d_in[0];
  const float* w = (const float*)d_in[1];
  const float* b = (const float*)d_in[2];
  float* out = (float*)d_out;
  const int n = in_sizes[0];

  // 128 elements per wave-iteration; aim for ~4-8 iterations per wave.
  const int  block  = 256;                       // 8 waves per block (wave32)
  long chunks = ((long)n + 127) / 128;
  long blocks = (chunks + 8 * 8 - 1) / (8 * 8);  // 8 waves/block * ~8 iters/wave
  if (blocks < 1)    blocks = 1;
  if (blocks > 4096) blocks = 4096;

  PolyModel_wmma_kernel<<<(int)blocks, block, 0, stream>>>(X, w, b, out, n);
}